// GAIN_BERT_48576080118242
// MI455X (gfx1250) — compile-verified
//
#include <hip/hip_runtime.h>
#include <hip/hip_bf16.h>

typedef float v2f __attribute__((ext_vector_type(2)));
typedef float v8f __attribute__((ext_vector_type(8)));

namespace {
constexpr int kB = 64, kSLEN = 512, kHID = 768, kTD = 32;
constexpr int kS = 15, kT = 40, kNPD = 1 + kS + kT;   // 56
constexpr int kN = kB * kNPD;                          // 3584
constexpr int kE = 100000;
constexpr int kGIN = kHID + kTD;                       // 800
constexpr int kGH = 512, kGO = 256;
constexpr int kLD = 512, kNC = 5;
constexpr int kBANK = kGIN + kGH + kGO;                // 1568

// workspace layout (float offsets)
constexpr size_t OFF_NODES = 0;
constexpr size_t SZ_NODES  = (size_t)kN * kGIN;         // 2,867,200
constexpr size_t OFF_AGG1  = OFF_NODES + SZ_NODES;
constexpr size_t SZ_AGG1   = (size_t)kN * 3 * kGIN;     // 8,601,600
constexpr size_t OFF_AGG2  = OFF_AGG1 + SZ_AGG1;
constexpr size_t SZ_AGG2   = (size_t)kN * 3 * kGH;      // 5,505,024
constexpr size_t OFF_DEG   = OFF_AGG2 + SZ_AGG2;
constexpr size_t SZ_DEG    = (size_t)3 * kN;
constexpr size_t OFF_CNT   = OFF_DEG + SZ_DEG;
constexpr size_t SZ_CNT    = (size_t)kB * (kS + kT);
constexpr size_t ZERO_FLOATS = OFF_CNT + SZ_CNT;        // zero-init region
constexpr size_t OFF_F1  = ZERO_FLOATS;
constexpr size_t SZ_F1   = (size_t)kN * kGH;
constexpr size_t OFF_F2  = OFF_F1 + SZ_F1;
constexpr size_t SZ_F2   = (size_t)kN * kGO;
constexpr size_t OFF_DOC = OFF_F2 + SZ_F2;
constexpr size_t SZ_DOC  = (size_t)kB * kBANK;
constexpr size_t OFF_HID = OFF_DOC + SZ_DOC;
} // namespace

// ---------------- zero accumulator region (float4 grid stride) ----------------
__global__ void gain_zero_kernel(float4* p, size_t n4) {
    size_t i = (size_t)blockIdx.x * blockDim.x + threadIdx.x;
    size_t stride = (size_t)gridDim.x * blockDim.x;
    float4 z = make_float4(0.f, 0.f, 0.f, 0.f);
    for (; i < n4; i += stride) p[i] = z;
}

// ---------------- scatter mean-pool: one block per (b,l) row ----------------
__global__ void gain_scatter_pool(const float* __restrict__ enc,
                                  const int* __restrict__ sid,
                                  const int* __restrict__ tid_,
                                  float* __restrict__ nodes,
                                  float* __restrict__ cnt) {
    int row = blockIdx.x;                 // b*SLEN + l
    int b = row / kSLEN;
    int s = sid[row], t = tid_[row];
    const float* src = enc + (size_t)row * kHID;
    if (s > 0) {
        float* dstp = nodes + (size_t)(b * kNPD + 1 + (s - 1)) * kGIN;
        for (int d = threadIdx.x; d < kHID; d += blockDim.x)
            atomicAdd(&dstp[d], src[d]);
        if (threadIdx.x == 0) atomicAdd(&cnt[b * (kS + kT) + (s - 1)], 1.0f);
    }
    if (t > 0) {
        float* dstp = nodes + (size_t)(b * kNPD + 1 + kS + (t - 1)) * kGIN;
        for (int d = threadIdx.x; d < kHID; d += blockDim.x)
            atomicAdd(&dstp[d], src[d]);
        if (threadIdx.x == 0) atomicAdd(&cnt[b * (kS + kT) + kS + (t - 1)], 1.0f);
    }
}

// ---------------- normalize pooled rows, add type embed, fill doc rows --------
__global__ void gain_finalize_nodes(const float* __restrict__ sentence_cls,
                                    const float* __restrict__ type_embed,
                                    const float* __restrict__ cnt,
                                    float* __restrict__ nodes) {
    int n = blockIdx.x;                    // node id
    int b = n / kNPD, k = n % kNPD;
    float* rowp = nodes + (size_t)n * kGIN;
    if (k == 0) {
        const float* cls = sentence_cls + (size_t)b * kHID;
        for (int d = threadIdx.x; d < kHID; d += blockDim.x) rowp[d] = cls[d];
        for (int d = threadIdx.x; d < kTD; d += blockDim.x)
            rowp[kHID + d] = type_embed[d];
    } else {
        int te;
        float c;
        if (k <= kS) { c = cnt[b * (kS + kT) + (k - 1)]; te = 1; }
        else         { c = cnt[b * (kS + kT) + kS + (k - 1 - kS)]; te = 2; }
        float inv = 1.0f / fmaxf(c, 1.0f);
        for (int d = threadIdx.x; d < kHID; d += blockDim.x) rowp[d] *= inv;
        for (int d = threadIdx.x; d < kTD; d += blockDim.x)
            rowp[kHID + d] = type_embed[te * kTD + d];
    }
}

// ---------------- per-relation in-degree ----------------
__global__ void gain_degree(const int* __restrict__ d0, const int* __restrict__ d1,
                            const int* __restrict__ d2, float* __restrict__ deg) {
    int idx = blockIdx.x * blockDim.x + threadIdx.x;
    if (idx >= 3 * kE) return;
    int r = idx / kE, e = idx - r * kE;
    const int* dp = (r == 0) ? d0 : (r == 1) ? d1 : d2;
    atomicAdd(&deg[r * kN + dp[e]], 1.0f);
}

// ---------------- edge scatter with mean normalization ----------------
// agg row stride = 3*F ; relation r occupies columns [r*F, (r+1)*F)
__global__ void gain_scatter_agg(const float* __restrict__ x, int F,
                                 const int* __restrict__ s0, const int* __restrict__ d0,
                                 const int* __restrict__ s1, const int* __restrict__ d1,
                                 const int* __restrict__ s2, const int* __restrict__ d2,
                                 const float* __restrict__ deg,
                                 float* __restrict__ agg) {
    int gid = blockIdx.x;                  // r*E + e
    int r = gid / kE, e = gid - r * kE;
    const int* sp = (r == 0) ? s0 : (r == 1) ? s1 : s2;
    const int* dp = (r == 0) ? d0 : (r == 1) ? d1 : d2;
    int s = sp[e], d = dp[e];
    float inv = 1.0f / fmaxf(deg[r * kN + d], 1.0f);
    const float* xr = x + (size_t)s * F;
    float* ar = agg + (size_t)d * (3 * F) + r * F;
    for (int j = threadIdx.x; j < F; j += blockDim.x)
        atomicAdd(&ar[j], xr[j] * inv);
}

// ---------------- FP32 WMMA GEMM: C = act(A[M,K] @ B[K,Nc] + bias) ----------
// blockDim = 32 (one wave, EXEC all-ones). One wave -> 16(M) x 64(N) tile.
// grid.x = (M/16) * (Nc/64).  K % 4 == 0, Nc % 64 == 0, M % 16 == 0.
__global__ void gain_wmma_gemm(const float* __restrict__ A,
                               const float* __restrict__ B,
                               const float* __restrict__ bias,
                               float* __restrict__ C,
                               int M, int K, int Nc, int relu) {
    int ntiles = Nc >> 6;
    int mt = blockIdx.x / ntiles;
    int nt = blockIdx.x - mt * ntiles;
    int lane = threadIdx.x & 31;
    int m0 = mt << 4;
    int n0 = nt << 6;

    int aRow  = m0 + (lane & 15);
    int kHalf = (lane >> 4) << 1;           // 0 or 2
    int bCol  = lane & 15;

    v8f acc0 = {}, acc1 = {}, acc2 = {}, acc3 = {};
    const float* Arow = A + (size_t)aRow * K;

    for (int k = 0; k < K; k += 4) {
        // A tile 16x4: lanes 0-15 hold K=k..k+1, lanes 16-31 hold K=k+2..k+3
        v2f a = *(const v2f*)(Arow + k + kHalf);
        const float* Br0 = B + (size_t)(k + kHalf) * Nc + n0 + bCol;
        const float* Br1 = Br0 + Nc;
        v2f b0, b1, b2, b3;
        b0[0] = Br0[0];  b0[1] = Br1[0];
        b1[0] = Br0[16]; b1[1] = Br1[16];
        b2[0] = Br0[32]; b2[1] = Br1[32];
        b3[0] = Br0[48]; b3[1] = Br1[48];
        acc0 = __builtin_amdgcn_wmma_f32_16x16x4_f32(false, a, false, b0, (short)0, acc0, false, false);
        acc1 = __builtin_amdgcn_wmma_f32_16x16x4_f32(false, a, false, b1, (short)0, acc1, false, false);
        acc2 = __builtin_amdgcn_wmma_f32_16x16x4_f32(false, a, false, b2, (short)0, acc2, false, false);
        acc3 = __builtin_amdgcn_wmma_f32_16x16x4_f32(false, a, false, b3, (short)0, acc3, false, false);
    }

    // C/D layout: vgpr v, lanes 0-15 -> M = v, lanes 16-31 -> M = v+8; N = lane%16
    int rBase = m0 + ((lane >> 4) << 3);
    int c0 = n0 + (lane & 15);
    for (int v = 0; v < 8; ++v) {
        int row = rBase + v;
        float x0 = acc0[v] + bias[c0];
        float x1 = acc1[v] + bias[c0 + 16];
        float x2 = acc2[v] + bias[c0 + 32];
        float x3 = acc3[v] + bias[c0 + 48];
        if (relu) {
            x0 = fmaxf(x0, 0.f); x1 = fmaxf(x1, 0.f);
            x2 = fmaxf(x2, 0.f); x3 = fmaxf(x3, 0.f);
        }
        float* Cr = C + (size_t)row * Nc;
        Cr[c0] = x0; Cr[c0 + 16] = x1; Cr[c0 + 32] = x2; Cr[c0 + 48] = x3;
    }
}

// ---------------- gather doc rows: [nodes | f1 | f2] -> doc_feat [64,1568] ----
__global__ void gain_doc_feat(const float* __restrict__ nodes,
                              const float* __restrict__ f1,
                              const float* __restrict__ f2,
                              float* __restrict__ doc) {
    int b = blockIdx.x;
    int n = b * kNPD;
    float* o = doc + (size_t)b * kBANK;
    const float* nr = nodes + (size_t)n * kGIN;
    const float* f1r = f1 + (size_t)n * kGH;
    const float* f2r = f2 + (size_t)n * kGO;
    for (int d = threadIdx.x; d < kGIN; d += blockDim.x) o[d] = nr[d];
    for (int d = threadIdx.x; d < kGH;  d += blockDim.x) o[kGIN + d] = f1r[d];
    for (int d = threadIdx.x; d < kGO;  d += blockDim.x) o[kGIN + kGH + d] = f2r[d];
}

// ---------------- final tiny GEMV: out[64,5] = hid @ Wp2 + bp2 ----------------
__global__ void gain_final(const float* __restrict__ hid,
                           const float* __restrict__ Wp2,
                           const float* __restrict__ bp2,
                           float* __restrict__ out) {
    int i = blockIdx.x * blockDim.x + threadIdx.x;
    if (i >= kB * kNC) return;
    int b = i / kNC, c = i - b * kNC;
    const float* h = hid + (size_t)b * kLD;
    float acc = bp2[c];
    for (int j = 0; j < kLD; ++j) acc += h[j] * Wp2[j * kNC + c];
    out[i] = acc;
}

extern "C" void kernel_launch(void* const* d_in, const int* in_sizes, int n_in,
                              void* d_out, int out_size, void* d_ws, size_t ws_size,
                              hipStream_t stream) {
    const float* enc   = (const float*)d_in[0];
    const float* cls   = (const float*)d_in[1];
    const int*   sid   = (const int*)d_in[2];
    const int*   tid_  = (const int*)d_in[3];
    const int* sG = (const int*)d_in[4];
    const int* dG = (const int*)d_in[5];
    const int* sS = (const int*)d_in[6];
    const int* dS = (const int*)d_in[7];
    const int* sT = (const int*)d_in[8];
    const int* dT = (const int*)d_in[9];
    const float* type_embed = (const float*)d_in[10];
    const float* W1  = (const float*)d_in[11];   // [3,800,512] == [2400,512]
    const float* b1  = (const float*)d_in[12];
    const float* W2  = (const float*)d_in[13];   // [3,512,256] == [1536,256]
    const float* b2  = (const float*)d_in[14];
    const float* Wp1 = (const float*)d_in[15];   // [1568,512]
    const float* bp1 = (const float*)d_in[16];
    const float* Wp2 = (const float*)d_in[17];   // [512,5]
    const float* bp2 = (const float*)d_in[18];
    float* out = (float*)d_out;

    float* ws = (float*)d_ws;
    float* nodes = ws + OFF_NODES;
    float* agg1  = ws + OFF_AGG1;
    float* agg2  = ws + OFF_AGG2;
    float* deg   = ws + OFF_DEG;
    float* cnt   = ws + OFF_CNT;
    float* f1    = ws + OFF_F1;
    float* f2    = ws + OFF_F2;
    float* doc   = ws + OFF_DOC;
    float* hid   = ws + OFF_HID;

    // 1) zero accumulators (nodes, agg1, agg2, deg, cnt are contiguous)
    gain_zero_kernel<<<2048, 256, 0, stream>>>((float4*)ws, ZERO_FLOATS / 4);

    // 2) scatter mean-pool contributions
    gain_scatter_pool<<<kB * kSLEN, 256, 0, stream>>>(enc, sid, tid_, nodes, cnt);

    // 3) normalize + type embeddings + doc rows
    gain_finalize_nodes<<<kN, 256, 0, stream>>>(cls, type_embed, cnt, nodes);

    // 4) degrees per relation
    gain_degree<<<(3 * kE + 255) / 256, 256, 0, stream>>>(dG, dS, dT, deg);

    // 5) layer-1 aggregation: agg1[N, 3*800]
    gain_scatter_agg<<<3 * kE, 128, 0, stream>>>(nodes, kGIN, sG, dG, sS, dS, sT, dT,
                                                 deg, agg1);

    // 6) f1 = relu(agg1 @ W1cat + b1), [3584,512]
    gain_wmma_gemm<<<(kN / 16) * (kGH / 64), 32, 0, stream>>>(
        agg1, W1, b1, f1, kN, 3 * kGIN, kGH, 1);

    // 7) layer-2 aggregation: agg2[N, 3*512]
    gain_scatter_agg<<<3 * kE, 128, 0, stream>>>(f1, kGH, sG, dG, sS, dS, sT, dT,
                                                 deg, agg2);

    // 8) f2 = relu(agg2 @ W2cat + b2), [3584,256]
    gain_wmma_gemm<<<(kN / 16) * (kGO / 64), 32, 0, stream>>>(
        agg2, W2, b2, f2, kN, 3 * kGH, kGO, 1);

    // 9) doc_feat = [nodes | f1 | f2] at doc rows, [64,1568]
    gain_doc_feat<<<kB, 256, 0, stream>>>(nodes, f1, f2, doc);

    // 10) hid = relu(doc_feat @ Wp1 + bp1), [64,512]
    gain_wmma_gemm<<<(kB / 16) * (kLD / 64), 32, 0, stream>>>(
        doc, Wp1, bp1, hid, kB, kBANK, kLD, 1);

    // 11) out = hid @ Wp2 + bp2, [64,5]
    gain_final<<<(kB * kNC + 255) / 256, 256, 0, stream>>>(hid, Wp2, bp2, out);
}